// SetFeatureAbstraction_26731876450578
// MI455X (gfx1250) — compile-verified
//
#include <hip/hip_runtime.h>
#include <math.h>

// Problem constants (from the reference)
#define B_  8
#define N_  8192
#define F_  256
#define M_  (N_ / 2)      // 4096
#define T_  1024          // threads per FPS block (32 waves of 32)
#define PPT (N_ / T_)     // 8 contiguous points per thread

// Dynamic LDS layout for fps_kernel (dynamic base = LDS offset 0: no static LDS)
// [0, 96KB)   float pts[N_*3]          (AoS copy of this batch's spatial points)
// [96KB, +32) u64 slots[4]             (rotating argmax reduction slots)
#define LDS_PTS_BYTES (N_ * 3 * 4)                 // 98304
#define LDS_SLOT_OFF  (LDS_PTS_BYTES)
#define LDS_TOTAL     (LDS_SLOT_OFF + 4 * 8)

// Sortable key: dist is >= 0 (or +inf), so its IEEE bits are monotonic as u32.
// ~index in the low word => max-key == (max value, lowest index on ties),
// matching jnp.argmax first-occurrence tie-breaking exactly.
__device__ __forceinline__ unsigned long long pack_key(float v, int idx)
{
    return ((unsigned long long)__float_as_uint(v) << 32) |
           (unsigned int)(~idx);
}

// ---------------------------------------------------------------------------
// FPS: one workgroup per batch element. Spatial -> LDS via async B128 copies,
// per-thread running distances in registers, packed-u64 argmax with a single
// barrier per step (wave32 shuffle reduce + one ds_max_u64 per wave).
// ---------------------------------------------------------------------------
extern "C" __global__ void __launch_bounds__(T_, 1)
fps_kernel(const float* __restrict__ spatial, int* __restrict__ idx_out)
{
    extern __shared__ char smem[];
    float*              pts   = (float*)smem;                       // [N_*3]
    unsigned long long* slots = (unsigned long long*)(smem + LDS_SLOT_OFF);

    const int b   = blockIdx.x;
    const int tid = threadIdx.x;
    const float* sp = spatial + (size_t)b * N_ * 3;

    // ---- async copy: this batch's 96KB of spatial data -> LDS (AoS) ----
    // Each thread moves its 8 contiguous points (96 bytes) as 6 x b128.
    {
        const char*  gbase  = (const char*)sp + (size_t)tid * (PPT * 3 * 4);
        unsigned int ldsoff = (unsigned int)tid * (PPT * 3 * 4);
#pragma unroll
        for (int j = 0; j < 6; ++j) {
            const void*  g = (const void*)(gbase + j * 16);
            unsigned int l = ldsoff + (unsigned int)(j * 16);
            asm volatile("global_load_async_to_lds_b128 %0, %1, off"
                         :: "v"(l), "v"(g) : "memory");
        }
    }

    // ---- register-resident copy of this thread's 8 points + distances ----
    float px[PPT], py[PPT], pz[PPT], dc[PPT];
    {
        const float* gp = sp + (size_t)tid * (PPT * 3);
#pragma unroll
        for (int k = 0; k < PPT; ++k) {
            px[k] = gp[3 * k + 0];
            py[k] = gp[3 * k + 1];
            pz[k] = gp[3 * k + 2];
            dc[k] = __builtin_inff();
        }
    }

    if (tid < 4) slots[tid] = 0ull;

    asm volatile("s_wait_asynccnt 0" ::: "memory");
    __syncthreads();

    if (tid == 0) idx_out[(size_t)b * M_ + 0] = 0;

    int sel = 0;
    const int base_idx = tid * PPT;   // contiguous ownership -> exact tie-break
    const int lane     = tid & 31;

    for (int s = 1; s < M_; ++s) {
        // Broadcast read of the selected point (same LDS address, no conflict)
        const float qx = pts[3 * sel + 0];
        const float qy = pts[3 * sel + 1];
        const float qz = pts[3 * sel + 2];

        unsigned long long bk = 0ull;
#pragma unroll
        for (int k = 0; k < PPT; ++k) {
            const float dx = px[k] - qx;
            const float dy = py[k] - qy;
            const float dz = pz[k] - qz;
            const float nd = fminf(dc[k], dx * dx + dy * dy + dz * dz);
            dc[k] = nd;
            const unsigned long long key = pack_key(nd, base_idx + k);
            bk = (key > bk) ? key : bk;
        }

        // wave32 argmax reduce on the packed key
#pragma unroll
        for (int off = 16; off > 0; off >>= 1) {
            const unsigned long long ok = __shfl_down(bk, off, 32);
            bk = (ok > bk) ? ok : bk;
        }

        // one ds_max_u64 per wave into the rotating slot for this step
        if (lane == 0) atomicMax(&slots[s & 3], bk);
        // reset the slot that will be used two steps from now; race-free:
        // its readers finished before barrier(s-1), its next writers start
        // after barrier(s+1).
        if (tid == 0) slots[(s + 2) & 3] = 0ull;

        __syncthreads();

        const unsigned long long win = slots[s & 3];
        sel = (int)(~(unsigned int)win);
        if (tid == 0) idx_out[(size_t)b * M_ + s] = sel;
    }
}

// ---------------------------------------------------------------------------
// Gather: out[b, m, :] = x[b, idx[b, m], :].  Pure bandwidth, B128 moves.
// 256 threads/block, 4 feature rows per block (64 float4 per row).
// ---------------------------------------------------------------------------
extern "C" __global__ void __launch_bounds__(256)
gather_kernel(const float* __restrict__ x, const int* __restrict__ idx,
              float* __restrict__ out)
{
    const int r = blockIdx.x * 4 + (threadIdx.x >> 6);   // flat row in [0, B*M)
    const int c = threadIdx.x & 63;                      // float4 column
    const int b = r / M_;
    const int i = idx[r];                                // idx is flat [B][M]

    const float4* src = (const float4*)(x + ((size_t)b * N_ + (size_t)i) * F_);
    float4*       dst = (float4*)(out + (size_t)r * F_);
    dst[c] = src[c];
}

// ---------------------------------------------------------------------------
extern "C" void kernel_launch(void* const* d_in, const int* in_sizes, int n_in,
                              void* d_out, int out_size, void* d_ws, size_t ws_size,
                              hipStream_t stream)
{
    (void)in_sizes; (void)n_in; (void)out_size; (void)ws_size;
    const float* x       = (const float*)d_in[0];   // [B, N, F] f32
    const float* spatial = (const float*)d_in[1];   // [B, N, D] f32
    float*       out     = (float*)d_out;           // [B, M, F] f32
    int*         idx     = (int*)d_ws;              // [B, M] i32 scratch

    fps_kernel<<<B_, T_, LDS_TOTAL, stream>>>(spatial, idx);
    gather_kernel<<<(B_ * M_) / 4, 256, 0, stream>>>(x, idx, out);
}